// HyperLayer_55155970015946
// MI455X (gfx1250) — compile-verified
//
#include <hip/hip_runtime.h>
#include <stdint.h>

// ---------------- problem constants (mirror reference) ----------------
#define BATCH    16
#define KTUP     16384
#define ADDS     32
#define NCAND    36          // 2^RANK + ADD = 4 + 32
#define IN_DIM   65536
#define OUT_DIM  65536
#define EPSF     1e-6f
#define NSPLIT   8           // K-chunks per batch row -> 128 workgroups
#define KC       (KTUP / NSPLIT)   // 2048 tuples per workgroup
#define THREADS  1024        // 32 wave32 waves
#define HALF_N   8388608u    // (B*K*ADD*RANK)/2 : threefry lane-pair distance

// CDNA5 async global->LDS path (guarded; falls back to plain copies)
#if defined(__HIP_DEVICE_COMPILE__) && \
    __has_builtin(__builtin_amdgcn_global_load_async_to_lds_b128) && \
    __has_builtin(__builtin_amdgcn_s_wait_asynccnt)
#define USE_ASYNC_LDS 1
#else
#define USE_ASYNC_LDS 0
#endif

typedef int v4i __attribute__((ext_vector_type(4)));

// ---------------- threefry2x32 with key = jax.random.key(42) = {0,42} ----
__device__ __forceinline__ uint2 threefry2x32_k42(uint32_t c0, uint32_t c1) {
  const uint32_t ks0 = 0u;
  const uint32_t ks1 = 42u;
  const uint32_t ks2 = 0x1BD11BDAu ^ 42u;
  uint32_t x0 = c0 + ks0;
  uint32_t x1 = c1 + ks1;
#define TF_R(d) { x0 += x1; x1 = (x1 << (d)) | (x1 >> (32 - (d))); x1 ^= x0; }
  TF_R(13) TF_R(15) TF_R(26) TF_R(6)   x0 += ks1; x1 += ks2 + 1u;
  TF_R(17) TF_R(29) TF_R(16) TF_R(24)  x0 += ks2; x1 += ks0 + 2u;
  TF_R(13) TF_R(15) TF_R(26) TF_R(6)   x0 += ks0; x1 += ks1 + 3u;
  TF_R(17) TF_R(29) TF_R(16) TF_R(24)  x0 += ks1; x1 += ks2 + 4u;
  TF_R(13) TF_R(15) TF_R(26) TF_R(6)   x0 += ks2; x1 += ks0 + 5u;
#undef TF_R
  return make_uint2(x0, x1);
}

// jax.random.uniform element at flat counter index idx (counts = iota(N),
// split into halves: out[i] pairs with out[i+N/2] from lanes (i, i+N/2))
__device__ __forceinline__ float tf_uniform(uint32_t idx) {
  uint32_t bits;
  if (idx < HALF_N) {
    bits = threefry2x32_k42(idx, idx + HALF_N).x;
  } else {
    bits = threefry2x32_k42(idx - HALF_N, idx).y;
  }
  // bits -> float in [0,1): ((bits>>9)|0x3f800000) as float, minus 1
  return __uint_as_float((bits >> 9) | 0x3f800000u) - 1.0f;
}

// ---------------- kernel: gather/scatter with LDS-privatized output ------
// (defined first so the disasm snippet shows the async global->LDS prologue)
__global__ __launch_bounds__(THREADS) void hyper_scatter_kernel(
    const float* __restrict__ x,
    const float* __restrict__ res,
    float* __restrict__ y) {
  extern __shared__ float smem[];
  float* acc  = smem;                // OUT_DIM floats  (256 KB) accumulator
  float* sres = smem + OUT_DIM;      // KC*4 floats     (32 KB)  res staging

  const int b     = blockIdx.x / NSPLIT;
  const int chunk = blockIdx.x - b * NSPLIT;
  const int k0    = chunk * KC;
  const int tid   = threadIdx.x;

  const float4* __restrict__ gres =
      reinterpret_cast<const float4*>(res) + (size_t)b * KTUP + k0;

  // ---- stage this chunk's res rows into LDS (async on CDNA5) ----
#if USE_ASYNC_LDS
  for (int i = tid; i < KC; i += THREADS) {
    __builtin_amdgcn_global_load_async_to_lds_b128(
        (__attribute__((address_space(1))) v4i*)(gres + i),
        (__attribute__((address_space(3))) v4i*)(sres + 4 * i),
        0, 0);
  }
#else
  for (int i = tid; i < KC; i += THREADS)
    reinterpret_cast<float4*>(sres)[i] = gres[i];
#endif

  // ---- zero the LDS accumulator while the async copies are in flight ----
  const float4 z4 = make_float4(0.f, 0.f, 0.f, 0.f);
  float4* acc4 = reinterpret_cast<float4*>(acc);
  for (int i = tid; i < OUT_DIM / 4; i += THREADS) acc4[i] = z4;

#if USE_ASYNC_LDS
  __builtin_amdgcn_s_wait_asynccnt(0);
#endif
  __syncthreads();

  const float* __restrict__ xb = x + (size_t)b * IN_DIM;

  for (int t = tid; t < KC; t += THREADS) {
    const int k = k0 + t;
    const float4 r = reinterpret_cast<const float4*>(sres)[t];

    // means in [0, 65535], shared isotropic sigma scaled by 65536
    const float m0 = 65535.0f / (1.0f + __expf(-r.x));
    const float m1 = 65535.0f / (1.0f + __expf(-r.y));
    const float sb = r.z + 2.0f;                       // + SIGMA_BOOST
    const float sp = fmaxf(sb, 0.0f) + log1pf(__expf(-fabsf(sb)));
    const float sig = (sp + EPSF) * 65536.0f;
    const float inv = 1.0f / sig;
    const float val = r.w;

    auto gauss = [&](float i0, float i1) -> float {
      const float z0 = (i0 - m0) * inv;
      const float z1 = (i1 - m1) * inv;
      return __expf(-0.5f * (z0 * z0 + z1 * z1));
    };

    uint32_t cand[NCAND];   // out_idx | (in_idx << 16), both < 65536
    float    prop[NCAND];

    // 4 floor/ceil corner candidates
    const float f0 = floorf(m0), c0 = ceilf(m0);
    const float f1 = floorf(m1), c1 = ceilf(m1);
    cand[0] = (uint32_t)(int)f0 | ((uint32_t)(int)f1 << 16); prop[0] = gauss(f0, f1);
    cand[1] = (uint32_t)(int)f0 | ((uint32_t)(int)c1 << 16); prop[1] = gauss(f0, c1);
    cand[2] = (uint32_t)(int)c0 | ((uint32_t)(int)f1 << 16); prop[2] = gauss(c0, f1);
    cand[3] = (uint32_t)(int)c0 | ((uint32_t)(int)c1 << 16); prop[3] = gauss(c0, c1);

    // 32 sampled candidates from the exact jax threefry stream
    const uint32_t base = (((uint32_t)b * KTUP + (uint32_t)k) * ADDS) * 2u;
#pragma unroll
    for (int a = 0; a < ADDS; ++a) {
      const float u0 = tf_uniform(base + 2u * (uint32_t)a)      * (1.0f - EPSF);
      const float u1 = tf_uniform(base + 2u * (uint32_t)a + 1u) * (1.0f - EPSF);
      const float s0 = floorf(u0 * 65536.0f);
      const float s1 = floorf(u1 * 65536.0f);
      cand[4 + a] = (uint32_t)(int)s0 | ((uint32_t)(int)s1 << 16);
      prop[4 + a] = gauss(s0, s1);
    }

    float S = 0.0f;
#pragma unroll
    for (int p = 0; p < NCAND; ++p) S += prop[p];
    const float w = val / (S + (float)NCAND * EPSF);   // sum(prop + eps)

    // gather x, scatter into the LDS-resident output row (ds_add_f32)
#pragma unroll
    for (int p = 0; p < NCAND; ++p) {
      const uint32_t pk = cand[p];
      const int oi = (int)(pk & 0xFFFFu);
      const int ii = (int)(pk >> 16);
      atomicAdd(&acc[oi], w * prop[p] * xb[ii]);
    }
  }

  __syncthreads();

  // ---- flush: coalesced global atomics (NSPLIT chunks share a row) ----
  float* __restrict__ yb = y + (size_t)b * OUT_DIM;
  for (int i = tid; i < OUT_DIM; i += THREADS)
    atomicAdd(&yb[i], acc[i]);
}

// ---------------- kernel: y[b,o] = bias[o] -------------------------------
__global__ __launch_bounds__(256) void bias_init_kernel(
    const float* __restrict__ bias, float* __restrict__ y) {
  const int i = blockIdx.x * 256 + threadIdx.x;   // grid covers B*OUT_DIM
  y[i] = bias[i & (OUT_DIM - 1)];
}

// ---------------- launcher ----------------------------------------------
extern "C" void kernel_launch(void* const* d_in, const int* in_sizes, int n_in,
                              void* d_out, int out_size, void* d_ws, size_t ws_size,
                              hipStream_t stream) {
  (void)in_sizes; (void)n_in; (void)d_ws; (void)ws_size; (void)out_size;
  const float* x    = (const float*)d_in[0];
  const float* res  = (const float*)d_in[1];
  const float* bias = (const float*)d_in[2];
  // d_in[3] (temp_indices) is dead: the reference overwrites both columns.
  float* y = (float*)d_out;

  bias_init_kernel<<<(BATCH * OUT_DIM) / 256, 256, 0, stream>>>(bias, y);

  const size_t shmem = ((size_t)OUT_DIM + (size_t)KC * 4) * sizeof(float); // 288 KB
  hyper_scatter_kernel<<<BATCH * NSPLIT, THREADS, shmem, stream>>>(x, res, y);
}